// Sampler_79448305041877
// MI455X (gfx1250) — compile-verified
//
#include <hip/hip_runtime.h>
#include <hip/hip_bf16.h>
#include <cstdint>

#define TOK_B 64
#define TOK_L 64
#define TOK_V 50257
#define MASK_ID 50256
#define THRESH 0.9f
#define LOG2E_F 1.4426950408889634f

#define T1 256
#define CHUNK (T1 * 4)   // 1024 elements per chunk per array
#define NBUF 3           // async pipeline depth (chunks in flight)

// ---- CDNA5 split-counter waits (builtin if available, else inline asm) ----
#if defined(__has_builtin)
#if __has_builtin(__builtin_amdgcn_s_wait_asynccnt)
#define WAIT_ASYNC(n) __builtin_amdgcn_s_wait_asynccnt(n)
#endif
#if __has_builtin(__builtin_amdgcn_s_wait_dscnt)
#define WAIT_DS(n) __builtin_amdgcn_s_wait_dscnt(n)
#endif
#endif
#ifndef WAIT_ASYNC
#define WAIT_ASYNC(n) asm volatile("s_wait_asynccnt %0" ::"i"(n) : "memory")
#endif
#ifndef WAIT_DS
#define WAIT_DS(n) asm volatile("s_wait_dscnt %0" ::"i"(n) : "memory")
#endif

// Async global -> LDS copy, 16 bytes per lane, tracked by ASYNCcnt.
// Stream is touched exactly once (1.65 GB >> 192 MB L2): mark non-temporal.
__device__ __forceinline__ void async_copy_b128(const float* gsrc, uint32_t lds_off) {
  asm volatile("global_load_async_to_lds_b128 %0, %1, off th:TH_LOAD_NT"
               :: "v"(lds_off), "v"(gsrc)
               : "memory");
}

__device__ __forceinline__ uint32_t lds_offset_of(const void* p) {
  // Generic pointers to LDS carry the byte offset in the low 32 bits.
  return (uint32_t)(uintptr_t)p;
}

// Kernel 1: one block per (b,l) row. Fused gumbel-argmax + online log-sum-exp
// over V, streamed through LDS with triple-buffered async copies.
__global__ __launch_bounds__(T1) void sampler_row_kernel(
    const float* __restrict__ logits, const float* __restrict__ temps,
    const float* __restrict__ gumbel, float* __restrict__ out) {
  __shared__ __align__(16) float sbuf[NBUF][2][CHUNK];  // [buf][0=logit,1=gumbel]
  __shared__ float r_bz[T1];
  __shared__ int   r_bi[T1];
  __shared__ float r_bs[T1];
  __shared__ float r_m[T1];
  __shared__ float r_sum[T1];

  const int row = blockIdx.x;
  const int tid = threadIdx.x;
  // scaled*log2e in one multiply; argmax is invariant under the positive scale.
  const float kk = (1.0f / temps[row >> 6]) * LOG2E_F;

  const float* lrow = logits + (size_t)row * TOK_V;
  const float* grow = gumbel + (size_t)row * TOK_V;

  float best_z = -INFINITY, best_s = 0.0f;
  int   best_i = 0;
  float m = -INFINITY, sum = 0.0f;

  auto acc = [&](float lg, float gg, int i) {
    float s2 = lg * kk;                    // scaled logit in log2 domain
    float z  = fmaf(gg, LOG2E_F, s2);      // (scaled + gumbel) * log2e
    if (z > best_z) { best_z = z; best_i = i; best_s = s2; }
    float d = s2 - m;
    if (d > 0.0f) { sum = sum * __builtin_amdgcn_exp2f(-d) + 1.0f; m = s2; }
    else          { sum += __builtin_amdgcn_exp2f(d); }
  };

  // Row base is only 4B-aligned (V*4 % 16 != 0): peel to reach 16B alignment.
  const int pre = (4 - (row & 3)) & 3;
  if (tid < pre) acc(lrow[tid], grow[tid], tid);

  const int t4    = tid * 4;
  const int nfull = (TOK_V - pre) / CHUNK;  // 49 full chunks for all pre values

  auto issue = [&](int c, int buf) {
    const float* lp = lrow + pre + c * CHUNK + t4;
    const float* gp = grow + pre + c * CHUNK + t4;
    async_copy_b128(lp, lds_offset_of(&sbuf[buf][0][t4]));
    async_copy_b128(gp, lds_offset_of(&sbuf[buf][1][t4]));
  };

  if (nfull > 0) issue(0, 0);
  if (nfull > 1) issue(1, 1);
  if (nfull > 2) issue(2, 2);

  int buf = 0;
  for (int c = 0; c < nfull; ++c) {
    // Chunks c .. min(c+2, nfull-1) are in flight (2 async ops each, in-order).
    if      (c + 2 < nfull) { WAIT_ASYNC(4); }
    else if (c + 1 < nfull) { WAIT_ASYNC(2); }
    else                    { WAIT_ASYNC(0); }

    float4 lv = *(const float4*)&sbuf[buf][0][t4];
    float4 gv = *(const float4*)&sbuf[buf][1][t4];
    const int base = pre + c * CHUNK + t4;
    acc(lv.x, gv.x, base + 0);
    acc(lv.y, gv.y, base + 1);
    acc(lv.z, gv.z, base + 2);
    acc(lv.w, gv.w, base + 3);

    if (c + NBUF < nfull) {
      WAIT_DS(0);        // ds reads of this buffer must land before overwrite
      issue(c + NBUF, buf);
    }
    buf = (buf + 1 == NBUF) ? 0 : buf + 1;
  }

  // Tail (<= 81 elements), direct global loads.
  const int ti = pre + nfull * CHUNK + tid;
  if (ti < TOK_V) acc(lrow[ti], grow[ti], ti);

  // Block tree reduction: argmax (first-index tie-break) + log-sum-exp merge.
  r_bz[tid] = best_z; r_bi[tid] = best_i; r_bs[tid] = best_s;
  r_m[tid] = m; r_sum[tid] = sum;
  __syncthreads();
  for (int s = T1 / 2; s > 0; s >>= 1) {
    if (tid < s) {
      const int o = tid + s;
      float oz = r_bz[o]; int oi = r_bi[o];
      if (oz > r_bz[tid] || (oz == r_bz[tid] && oi < r_bi[tid])) {
        r_bz[tid] = oz; r_bi[tid] = oi; r_bs[tid] = r_bs[o];
      }
      float mo = r_m[o], mt = r_m[tid];
      float mn = fmaxf(mo, mt);
      r_sum[tid] = r_sum[tid] * __builtin_amdgcn_exp2f(mt - mn) +
                   r_sum[o]  * __builtin_amdgcn_exp2f(mo - mn);
      r_m[tid] = mn;
    }
    __syncthreads();
  }
  if (tid == 0) {
    const float lse2 = r_m[0] + __builtin_amdgcn_logf(r_sum[0]); // log2 LSE
    const float p    = __builtin_amdgcn_exp2f(r_bs[0] - lse2);   // x0_p
    out[2 * 4096 + row] = (float)r_bi[0];  // x0
    out[3 * 4096 + row] = p;               // x0_p
  }
}

__global__ void zero_kernel(float* __restrict__ out) { out[4 * 4096] = 0.0f; }

// Kernel 2: per-batch-row masking / dynamic-threshold transfer logic.
__global__ __launch_bounds__(TOK_L) void finalize_kernel(
    const int* __restrict__ x, float* __restrict__ out) {
  __shared__ float s_conf[TOK_L];
  __shared__ int   s_high[TOK_L];
  __shared__ int   s_mask[TOK_L];
  __shared__ int   sh_hasmask, sh_hashigh, sh_top1;

  const int b = blockIdx.x, l = threadIdx.x;
  const int row = b * TOK_L + l;
  const int   xv = x[row];
  const int   x0 = (int)out[2 * 4096 + row];
  const float p  = out[3 * 4096 + row];

  const bool  maskf = (xv == MASK_ID);
  const float conf  = maskf ? p : -INFINITY;
  const bool  high  = conf > THRESH;
  s_conf[l] = conf;
  s_high[l] = high ? 1 : 0;
  s_mask[l] = maskf ? 1 : 0;
  __syncthreads();

  if (l == 0) {
    int hm = 0, hh = 0, cnt = 0, t1 = 0;
    float best = -INFINITY;
    for (int i = 0; i < TOK_L; ++i) {
      hm |= s_mask[i];
      hh |= s_high[i];
      cnt += s_high[i];
      if (s_conf[i] > best) { best = s_conf[i]; t1 = i; }  // first max
    }
    sh_hasmask = hm; sh_hashigh = hh; sh_top1 = t1;
    const int c = hm ? (hh ? cnt : 1) : 0;
    if (c) atomicAdd(&out[4 * 4096], (float)c);  // exact small ints: deterministic
  }
  __syncthreads();

  const bool transfer =
      sh_hasmask && (sh_hashigh ? high : (l == sh_top1));
  out[row]        = (float)(transfer ? x0 : xv);  // x_new
  out[4096 + row] = transfer ? 1.0f : 0.0f;       // transfer
}

extern "C" void kernel_launch(void* const* d_in, const int* in_sizes, int n_in,
                              void* d_out, int out_size, void* d_ws, size_t ws_size,
                              hipStream_t stream) {
  (void)in_sizes; (void)n_in; (void)out_size; (void)d_ws; (void)ws_size;
  const float* logits = (const float*)d_in[0];
  const float* temps  = (const float*)d_in[1];
  const float* gumbel = (const float*)d_in[2];
  const int*   x      = (const int*)d_in[3];
  float* out = (float*)d_out;

  sampler_row_kernel<<<TOK_B * TOK_L, T1, 0, stream>>>(logits, temps, gumbel, out);
  zero_kernel<<<1, 1, 0, stream>>>(out);
  finalize_kernel<<<TOK_B, TOK_L, 0, stream>>>(x, out);
}